// CapsRoutingLayer_62036507623978
// MI455X (gfx1250) — compile-verified
//
#include <hip/hip_runtime.h>
#include <math.h>

// Problem constants (from reference setup_inputs)
#define B_SZ 128
#define NI   2048
#define DI   16
#define NO   32
#define DO   16
#define KSPLIT 8           // split-K factor for kernel 1
#define I_PER_KS (NI / KSPLIT)

typedef __attribute__((ext_vector_type(2))) float v2f;
typedef __attribute__((ext_vector_type(8))) float v8f;

__device__ __forceinline__ v8f wmma_f32_k4(v2f a, v2f b, v8f c) {
  // D = A(16x4 f32) * B(4x16 f32) + C(16x16 f32)
  return __builtin_amdgcn_wmma_f32_16x16x4_f32(
      /*neg_a=*/false, a, /*neg_b=*/false, b,
      /*c_mod=*/(short)0, c, /*reuse_a=*/false, /*reuse_b=*/false);
}

// ---------------------------------------------------------------------------
// Kernel 1: split-K partial of S0^T[n,b] = sum_{i,k} W[i,n,k]*X[b,i,k]
// grid = (32 n-tiles, 8 k-splits); block = 256 threads = 8 waves (wave = b-tile).
// Raw partial sums stored to s0part[ks][b][j][d].
// ---------------------------------------------------------------------------
__global__ __launch_bounds__(256) void caps_s0_kernel(
    const float* __restrict__ X, const float* __restrict__ W,
    float* __restrict__ s0part)
{
  const int nt   = blockIdx.x;          // j / n-tile
  const int ks   = blockIdx.y;          // k-split chunk
  const int wave = threadIdx.x >> 5;    // b-tile
  const int lane = threadIdx.x & 31;
  const int hl   = lane >> 4;           // half-wave (K split within WMMA)
  const int lm   = lane & 15;

  const int i0 = ks * I_PER_KS;
  // A fragment: row n = nt*16+lm of W[i] ([512][16] row-major), k = 2*hl+{0,1}+4s
  const float* wp = W + (size_t)i0 * 512 * DI + (size_t)(nt * 16 + lm) * DI + 2 * hl;
  // B fragment: col b = wave*16+lm, from X[b][i][k]
  const float* xp = X + (size_t)(wave * 16 + lm) * NI * DI + (size_t)i0 * DI + 2 * hl;

  v8f acc = {0.f, 0.f, 0.f, 0.f, 0.f, 0.f, 0.f, 0.f};
  for (int i = 0; i < I_PER_KS; ++i) {
    v2f a0 = *(const v2f*)(wp + 0);
    v2f a1 = *(const v2f*)(wp + 4);
    v2f a2 = *(const v2f*)(wp + 8);
    v2f a3 = *(const v2f*)(wp + 12);
    v2f b0 = *(const v2f*)(xp + 0);
    v2f b1 = *(const v2f*)(xp + 4);
    v2f b2 = *(const v2f*)(xp + 8);
    v2f b3 = *(const v2f*)(xp + 12);
    acc = wmma_f32_k4(a0, b0, acc);
    acc = wmma_f32_k4(a1, b1, acc);
    acc = wmma_f32_k4(a2, b2, acc);
    acc = wmma_f32_k4(a3, b3, acc);
    wp += 512 * 16;
    xp += 16;
  }

  // Tile: reg r, lane l -> (d = r + 8*hl, b = wave*16 + lm)
  float* pp = s0part
            + (((size_t)(ks * B_SZ + wave * 16 + lm)) * NO + nt) * DO + 8 * hl;
#pragma unroll
  for (int r = 0; r < 8; ++r) pp[r] = acc[r];
}

// ---------------------------------------------------------------------------
// Kernel 1b: deterministic 8-way split-K reduction, /32, squash -> v0[b][j][d]
// ---------------------------------------------------------------------------
__global__ __launch_bounds__(256) void caps_v0_kernel(
    const float* __restrict__ s0part, float* __restrict__ v0)
{
  int idx = blockIdx.x * 256 + threadIdx.x;   // b*NO + j
  if (idx >= B_SZ * NO) return;
  int b = idx >> 5;
  int j = idx & 31;

  float s[DO];
#pragma unroll
  for (int d = 0; d < DO; ++d) s[d] = 0.f;

  for (int ks = 0; ks < KSPLIT; ++ks) {
    const float* p = s0part + (((size_t)(ks * B_SZ + b)) * NO + j) * DO;
#pragma unroll
    for (int d = 0; d < DO; ++d) s[d] += p[d];
  }

  float n2 = 0.f;
#pragma unroll
  for (int d = 0; d < DO; ++d) { s[d] *= (1.0f / 32.0f); n2 += s[d] * s[d]; }
  float nn = sqrtf(n2);
  float scale = nn / (1.0f + n2);

  float* vp = v0 + (size_t)idx * DO;
#pragma unroll
  for (int d = 0; d < DO; ++d) vp[d] = s[d] * scale;
}

// ---------------------------------------------------------------------------
// Kernel 2: fused  b1 = <X_hat, v0>  -> softmax_j -> s1 partial accumulation.
// grid.x = 256 (bt = blk>>5, sub = blk&31), block = 128 threads = 4 waves.
// Xh tiles kept register-resident across the softmax (explicit xh[] array).
// Deterministic: barrier-ordered LDS merge, per-block partial to gpart.
// ---------------------------------------------------------------------------
__global__ __launch_bounds__(128) void caps_route_kernel(
    const float* __restrict__ X, const float* __restrict__ W,
    const float* __restrict__ v0, float* __restrict__ gpart)
{
  __shared__ float v0s[NO * 16 * DO];  // [j][b_local][d]  (32 KB)
  __shared__ float s1s[NO * 16 * DO];  // [j][b_local][d]  (32 KB)

  const int bt  = blockIdx.x >> 5;
  const int sub = blockIdx.x & 31;

  for (int t = threadIdx.x; t < NO * 16 * DO; t += 128) {
    int j  = t >> 8;
    int bl = (t >> 4) & 15;
    int d  = t & 15;
    v0s[t] = v0[(((size_t)(bt * 16 + bl)) * NO + j) * DO + d];
    s1s[t] = 0.f;
  }
  __syncthreads();

  const int wave = threadIdx.x >> 5;
  const int lane = threadIdx.x & 31;
  const int hl   = lane >> 4;
  const int lm   = lane & 15;

  v8f s1acc[NO];
#pragma unroll
  for (int j = 0; j < NO; ++j)
    s1acc[j] = (v8f){0.f, 0.f, 0.f, 0.f, 0.f, 0.f, 0.f, 0.f};

  const float* xbase = X + (size_t)(bt * 16 + lm) * NI * DI + 2 * hl;

  for (int t = 0; t < 16; ++t) {
    const int i = sub * 4 + wave + 128 * t;   // uniform across the wave

    const float* xp = xbase + (size_t)i * DI;
    v2f xb0 = *(const v2f*)(xp + 0);
    v2f xb1 = *(const v2f*)(xp + 4);
    v2f xb2 = *(const v2f*)(xp + 8);
    v2f xb3 = *(const v2f*)(xp + 12);

    const float* wbase = W + ((size_t)i * 512 + lm) * DI + 2 * hl;

    // Pass A: compute all 32 Xh tiles (register resident) and b1[j]
    v8f xh[NO];
    float b1[NO];
#pragma unroll
    for (int j = 0; j < NO; ++j) {
      const float* wp = wbase + (size_t)j * 16 * DI;
      v8f acc = {0.f, 0.f, 0.f, 0.f, 0.f, 0.f, 0.f, 0.f};
      acc = wmma_f32_k4(*(const v2f*)(wp + 0),  xb0, acc);
      acc = wmma_f32_k4(*(const v2f*)(wp + 4),  xb1, acc);
      acc = wmma_f32_k4(*(const v2f*)(wp + 8),  xb2, acc);
      acc = wmma_f32_k4(*(const v2f*)(wp + 12), xb3, acc);
      xh[j] = acc;
      // tile: reg r, lane l -> (d = r + 8*hl, b = lm); dot with v0[b][j][d]
      const float* vv = &v0s[(j * 16 + lm) * DO + 8 * hl];
      float t0 = 0.f;
#pragma unroll
      for (int r = 0; r < 8; ++r) t0 += acc[r] * vv[r];
      t0 += __shfl_xor(t0, 16);
      b1[j] = t0;
    }

    // lane-local softmax over j (value valid for b = lm, replicated per half)
    float m = b1[0];
#pragma unroll
    for (int j = 1; j < NO; ++j) m = fmaxf(m, b1[j]);
    float ssum = 0.f;
#pragma unroll
    for (int j = 0; j < NO; ++j) { b1[j] = __expf(b1[j] - m); ssum += b1[j]; }
    float inv = 1.0f / ssum;

    // Pass B: weighted accumulation of the register-resident tiles
#pragma unroll
    for (int j = 0; j < NO; ++j) {
      float c = b1[j] * inv;
#pragma unroll
      for (int r = 0; r < 8; ++r) s1acc[j][r] += c * xh[j][r];
    }
  }

  // Deterministic cross-wave merge (fixed order, no atomics)
  for (int k = 0; k < 4; ++k) {
    if (wave == k) {
#pragma unroll
      for (int j = 0; j < NO; ++j) {
#pragma unroll
        for (int r = 0; r < 8; ++r)
          s1s[(j * 16 + lm) * DO + 8 * hl + r] += s1acc[j][r];
      }
    }
    __syncthreads();
  }

  float* gp = gpart + (size_t)blockIdx.x * (NO * 16 * DO);
  for (int t = threadIdx.x; t < NO * 16 * DO; t += 128) gp[t] = s1s[t];
}

// ---------------------------------------------------------------------------
// Kernel 3: deterministic 32-way partial reduction + squash -> out[b][j][d]
// ---------------------------------------------------------------------------
__global__ __launch_bounds__(256) void caps_squash_out_kernel(
    const float* __restrict__ gpart, float* __restrict__ out)
{
  int idx = blockIdx.x * 256 + threadIdx.x;   // b*NO + j
  if (idx >= B_SZ * NO) return;
  int b  = idx >> 5;
  int j  = idx & 31;
  int bt = b >> 4;
  int bl = b & 15;

  float s[DO];
#pragma unroll
  for (int d = 0; d < DO; ++d) s[d] = 0.f;

  for (int sub = 0; sub < 32; ++sub) {
    const float* p = gpart + ((size_t)(bt * 32 + sub)) * (NO * 16 * DO)
                           + (j * 16 + bl) * DO;
#pragma unroll
    for (int d = 0; d < DO; ++d) s[d] += p[d];
  }

  float n2 = 0.f;
#pragma unroll
  for (int d = 0; d < DO; ++d) n2 += s[d] * s[d];
  float nn = sqrtf(n2);
  float scale = nn / (1.0f + n2);

  float* op = out + (size_t)idx * DO;
#pragma unroll
  for (int d = 0; d < DO; ++d) op[d] = s[d] * scale;
}

// ---------------------------------------------------------------------------
// Workspace layout (floats):
//   v0     : 65536                     (256 KB)
//   gpart  : 256 * 8192 = 2097152      (8 MB)
//   s0part : 8 * 65536  = 524288       (2 MB)
// total ~10.25 MB
// ---------------------------------------------------------------------------
extern "C" void kernel_launch(void* const* d_in, const int* in_sizes, int n_in,
                              void* d_out, int out_size, void* d_ws, size_t ws_size,
                              hipStream_t stream) {
  const float* X = (const float*)d_in[0];   // [128][2048][16]
  const float* W = (const float*)d_in[1];   // [2048][32][16][16]
  float* out = (float*)d_out;               // [128][32][16]

  float* v0     = (float*)d_ws;
  float* gpart  = v0 + (size_t)B_SZ * NO * DO;
  float* s0part = gpart + (size_t)256 * NO * 16 * DO;

  caps_s0_kernel<<<dim3(32, KSPLIT), dim3(256), 0, stream>>>(X, W, s0part);
  caps_v0_kernel<<<dim3(16), dim3(256), 0, stream>>>(s0part, v0);
  caps_route_kernel<<<dim3(256), dim3(128), 0, stream>>>(X, W, v0, gpart);
  caps_squash_out_kernel<<<dim3(16), dim3(256), 0, stream>>>(gpart, out);
}